// Cross_modal_ContrastiveLoss6_36618891166022
// MI455X (gfx1250) — compile-verified
//
#include <hip/hip_runtime.h>
#include <math.h>

#define N_SAMP 4096
#define D_DIM  2048
#define NCLS   128
#define MARGIN 0.5f

typedef __attribute__((ext_vector_type(2))) float v2f;
typedef __attribute__((ext_vector_type(8))) float v8f;

// ---------------------------------------------------------------- zero
__global__ void zero_kernel(float* ws, int nws, float* out, int nout) {
    int i = blockIdx.x * blockDim.x + threadIdx.x;
    if (i < nws)  ws[i]  = 0.0f;
    if (i < nout) out[i] = 0.0f;
}

// ---------------------------------------------------------------- class counts
__global__ void count_kernel(const int* __restrict__ targets, float* __restrict__ cnt) {
    int i = blockIdx.x * blockDim.x + threadIdx.x;
    if (i < N_SAMP) {
        int c = targets[i] & (NCLS - 1);
        atomicAdd(&cnt[c], 1.0f);
    }
}

// ---------------------------------------------------------------- segment sums
// One block owns a 32-column slice of D; accumulates all N rows for both
// modalities into LDS (128 classes x 32 cols x 2 modalities = 32KB), then
// writes its slice of the class-sum matrices with plain stores (no global
// atomics). grid = D/32 = 64 blocks, block = 256 threads (8 rows x 32 cols).
__global__ __launch_bounds__(256) void segsum_kernel(const float* __restrict__ m1,
                                                     const float* __restrict__ m2,
                                                     const int*   __restrict__ targets,
                                                     float* __restrict__ sumR,
                                                     float* __restrict__ sumT) {
    __shared__ float lsum[2][NCLS][32];   // 32 KB

    const int tid     = threadIdx.x;
    const int colBase = blockIdx.x * 32;
    const int col     = tid & 31;
    const int rsub    = tid >> 5;         // 0..7

    // zero LDS
    for (int i = tid; i < 2 * NCLS * 32; i += 256)
        ((float*)lsum)[i] = 0.0f;
    __syncthreads();

    for (int r0 = 0; r0 < N_SAMP; r0 += 8) {
        int row = r0 + rsub;
        int c   = targets[row] & (NCLS - 1);
        size_t idx = (size_t)row * D_DIM + colBase + col;
        atomicAdd(&lsum[0][c][col], m1[idx]);
        atomicAdd(&lsum[1][c][col], m2[idx]);
    }
    __syncthreads();

    // write out: each block owns its 32-column slice exclusively
    for (int i = tid; i < NCLS * 32; i += 256) {
        int c  = i >> 5;
        int cc = i & 31;
        sumR[(size_t)c * D_DIM + colBase + cc] = lsum[0][c][cc];
        sumT[(size_t)c * D_DIM + colBase + cc] = lsum[1][c][cc];
    }
}

// ---------------------------------------------------------------- means + norms
// In-place: sums -> means. Also row norms ||meanR||^2, ||meanT||^2,
// ||0.5(meanR+meanT)||^2. One block per class.
__global__ __launch_bounds__(256) void means_kernel(float* __restrict__ sumR,
                                                    float* __restrict__ sumT,
                                                    const float* __restrict__ cnt,
                                                    float* __restrict__ normR,
                                                    float* __restrict__ normT,
                                                    float* __restrict__ normC) {
    __shared__ float sR[256], sT[256], sC[256];
    const int c   = blockIdx.x;
    const int tid = threadIdx.x;
    const float inv = 1.0f / fmaxf(cnt[c], 1.0f);

    float aR = 0.0f, aT = 0.0f, aC = 0.0f;
    for (int d = tid; d < D_DIM; d += 256) {
        size_t idx = (size_t)c * D_DIM + d;
        float mR = sumR[idx] * inv;
        float mT = sumT[idx] * inv;
        sumR[idx] = mR;                       // now holds meanR
        sumT[idx] = mT;                       // now holds meanT
        float ctr = 0.5f * (mR + mT);
        aR += mR * mR;
        aT += mT * mT;
        aC += ctr * ctr;
    }
    sR[tid] = aR; sT[tid] = aT; sC[tid] = aC;
    __syncthreads();
    for (int s = 128; s > 0; s >>= 1) {
        if (tid < s) { sR[tid] += sR[tid+s]; sT[tid] += sT[tid+s]; sC[tid] += sC[tid+s]; }
        __syncthreads();
    }
    if (tid == 0) { normR[c] = sR[0]; normT[c] = sT[0]; normC[c] = sC[0]; }
}

// ---------------------------------------------------------------- WMMA distance GEMM + fused loss epilogue
// One wave32 per 16x16 output tile of the 128x128 class-pair matrix.
// G1 = meanR @ center^T, G2 = meanT @ center^T via V_WMMA_F32_16X16X4_F32,
// K = 2048 in steps of 4. B fragment (center) built on the fly.
//
// f32 16x16x4 fragment layout (ISA 7.12.2):
//   A (16x4):  lanes 0-15 -> M=lane, VGPR{0,1}=K{0,1}; lanes 16-31 -> K{2,3}
//   B (4x16):  lanes 0-15 -> N=lane, VGPR{0,1}=K{0,1}; lanes 16-31 -> K{2,3}
//   C (16x16): lane<16: N=lane, VGPR v -> M=v;  lane>=16: N=lane-16, M=v+8
__global__ __launch_bounds__(32) void wmma_loss_kernel(const float* __restrict__ meanR,
                                                       const float* __restrict__ meanT,
                                                       const float* __restrict__ cnt,
                                                       const float* __restrict__ normR,
                                                       const float* __restrict__ normT,
                                                       const float* __restrict__ normC,
                                                       float* __restrict__ out) {
    const int lane = threadIdx.x;
    const int bi   = blockIdx.x;     // 0..7  row tile (A classes)
    const int bj   = blockIdx.y;     // 0..7  col tile (B classes)

    const int rA   = bi * 16 + (lane & 15);
    const int rB   = bj * 16 + (lane & 15);
    const int ksub = (lane >> 4) * 2;

    const float* pRA = meanR + (size_t)rA * D_DIM + ksub;
    const float* pTA = meanT + (size_t)rA * D_DIM + ksub;
    const float* pRB = meanR + (size_t)rB * D_DIM + ksub;
    const float* pTB = meanT + (size_t)rB * D_DIM + ksub;

    v8f c1 = {};
    v8f c2 = {};

    for (int k = 0; k < D_DIM; k += 4) {
        v2f aR = *(const v2f*)(pRA + k);
        v2f aT = *(const v2f*)(pTA + k);
        v2f bR = *(const v2f*)(pRB + k);
        v2f bT = *(const v2f*)(pTB + k);
        v2f b  = (bR + bT) * 0.5f;           // center fragment
        c1 = __builtin_amdgcn_wmma_f32_16x16x4_f32(false, aR, false, b,
                                                   (short)0, c1, false, false);
        c2 = __builtin_amdgcn_wmma_f32_16x16x4_f32(false, aT, false, b,
                                                   (short)0, c2, false, false);
    }

    // Fused epilogue: squared dist -> clamp -> sqrt -> contrastive terms,
    // weighted by class-pair sample counts.
    const int mBase = (lane >> 4) * 8;
    const int n     = lane & 15;
    const int b     = bj * 16 + n;
    const float cb  = cnt[b];
    const float ncb = normC[b];

    float local = 0.0f;
    #pragma unroll
    for (int v = 0; v < 8; ++v) {
        int a = bi * 16 + mBase + v;
        float w = cnt[a] * cb;

        float sq1 = fmaxf(normR[a] + ncb - 2.0f * c1[v], 1e-12f);
        float d1  = sqrtf(sq1);
        float t1;
        if (a == b) t1 = sq1;                                 // label=1: dist^2
        else { float r = fmaxf(MARGIN - sqrtf(d1 + 1e-10f), 0.0f); t1 = r * r; }

        float sq2 = fmaxf(normT[a] + ncb - 2.0f * c2[v], 1e-12f);
        float d2  = sqrtf(sq2);
        float t2;
        if (a == b) t2 = sq2;
        else { float r = fmaxf(MARGIN - sqrtf(d2 + 1e-10f), 0.0f); t2 = r * r; }

        local += w * (t1 + t2);
    }

    // wave32 reduction
    for (int off = 16; off > 0; off >>= 1)
        local += __shfl_xor(local, off, 32);

    if (lane == 0)
        atomicAdd(out, local * (1.0f / ((float)N_SAMP * (float)N_SAMP)));
}

// ---------------------------------------------------------------- launch
extern "C" void kernel_launch(void* const* d_in, const int* in_sizes, int n_in,
                              void* d_out, int out_size, void* d_ws, size_t ws_size,
                              hipStream_t stream) {
    const float* m1      = (const float*)d_in[0];
    const float* m2      = (const float*)d_in[1];
    const int*   targets = (const int*)d_in[2];
    float*       out     = (float*)d_out;

    // workspace layout (floats)
    float* ws    = (float*)d_ws;
    float* sumR  = ws;                         // 128*2048 -> becomes meanR
    float* sumT  = sumR + NCLS * D_DIM;        // 128*2048 -> becomes meanT
    float* cnt   = sumT + NCLS * D_DIM;        // 128
    float* normR = cnt + NCLS;                 // 128
    float* normT = normR + NCLS;               // 128
    float* normC = normT + NCLS;               // 128

    const int nZero = 2 * NCLS * D_DIM + NCLS; // sums + counts
    zero_kernel<<<(nZero + 255) / 256, 256, 0, stream>>>(ws, nZero, out, out_size);

    count_kernel<<<(N_SAMP + 255) / 256, 256, 0, stream>>>(targets, cnt);

    segsum_kernel<<<D_DIM / 32, 256, 0, stream>>>(m1, m2, targets, sumR, sumT);

    means_kernel<<<NCLS, 256, 0, stream>>>(sumR, sumT, cnt, normR, normT, normC);

    wmma_loss_kernel<<<dim3(8, 8), 32, 0, stream>>>(sumR, sumT, cnt,
                                                    normR, normT, normC, out);
}